// GMLinearEdgeEncoder_74972949118977
// MI455X (gfx1250) — compile-verified
//
#include <hip/hip_runtime.h>
#include <hip/hip_bf16.h>

typedef __attribute__((ext_vector_type(2))) float v2f;
typedef __attribute__((ext_vector_type(8))) float v8f;

#define B_GRAPHS 64
#define N_NODES  128
#define EMB      16
#define OUT_DIM  64
#define E_REAL   65536
#define TOTAL    (B_GRAPHS * N_NODES * N_NODES)   // 1,048,576 pair slots

// ---------------------------------------------------------------------------
// Kernel 1: out_idx (as float; values <= 8191 are exact in f32)
// out_idx[0*TOTAL + i] = g*N + rem/N ; out_idx[1*TOTAL + i] = g*N + rem%N
// ---------------------------------------------------------------------------
__global__ void __launch_bounds__(256)
gm_idx_kernel(float* __restrict__ out_idx) {
  const int i   = blockIdx.x * 256 + threadIdx.x;     // grid covers TOTAL exactly
  const int g   = i >> 14;                            // / (N*N)
  const int rem = i & 16383;                          // % (N*N)
  out_idx[i]         = (float)(g * N_NODES + (rem >> 7));
  out_idx[TOTAL + i] = (float)(g * N_NODES + (rem & 127));
}

// ---------------------------------------------------------------------------
// Kernel 2: out_val = gm_val[1048576,16] @ W[64,16]^T via V_WMMA_F32_16X16X4_F32.
// One wave -> 16 rows x 64 cols. K=16 = 4 chained 16x16x4 WMMAs per N-tile.
//
// f32 WMMA operand layout (ISA 7.12.2):
//   A 16x4 : lane&15 = M row; half=lane>>4 selects K pair {4c,4c+1} / {4c+2,4c+3}
//   B 4x16 : lane&15 = N col; half selects same K pairs      -> float2 from W row
//   C/D    : 8 VGPRs; VGPR v -> row v+8*half, col lane&15
// ---------------------------------------------------------------------------
__global__ void __launch_bounds__(256)
gm_gemm_kernel(const float* __restrict__ gm_val,
               const float* __restrict__ W,
               float* __restrict__ out_val) {
  const int lane     = threadIdx.x & 31;
  const int wave     = blockIdx.x * 8 + (threadIdx.x >> 5);
  const int rowStart = wave * 16;            // 16 gm_val rows per wave
  const int half     = lane >> 4;            // 0: lanes 0-15, 1: lanes 16-31
  const int r        = lane & 15;

  // A-matrix chunks: a[c] = { A[r][4c+2h], A[r][4c+2h+1] }
  const float* arow = gm_val + (size_t)(rowStart + r) * EMB;
  v2f a[4];
#pragma unroll
  for (int c = 0; c < 4; ++c)
    a[c] = *(const v2f*)(arow + 4 * c + 2 * half);

#pragma unroll
  for (int n = 0; n < 4; ++n) {              // 4 tiles of 16 output columns
    // B[k][col] = W^T[k][16n+col] = W[16n+col][k]  (contiguous pair in W row)
    const float* wrow = W + (size_t)(16 * n + r) * EMB;
    v8f acc = {};
#pragma unroll
    for (int c = 0; c < 4; ++c) {
      v2f b = *(const v2f*)(wrow + 4 * c + 2 * half);
      acc = __builtin_amdgcn_wmma_f32_16x16x4_f32(
          /*neg_a=*/false, a[c], /*neg_b=*/false, b,
          /*c_mod=*/(short)0, acc, /*reuse_a=*/false, /*reuse_b=*/false);
    }
#pragma unroll
    for (int v = 0; v < 8; ++v) {
      const size_t idx = (size_t)(rowStart + v + 8 * half) * OUT_DIM + 16 * n + r;
      out_val[idx] = acc[v];
    }
  }
}

// ---------------------------------------------------------------------------
// Kernel 3: scatter-add real edges. Duplicate (src,dst) pairs are possible
// (random edges), so atomicAdd is required for correctness.
// ---------------------------------------------------------------------------
__global__ void __launch_bounds__(256)
gm_scatter_kernel(const int* __restrict__ edge_index,
                  const float* __restrict__ edge_attr,
                  float* __restrict__ out_val) {
  const int gid = blockIdx.x * 256 + threadIdx.x;   // grid covers E_REAL*OUT
  const int e   = gid >> 6;                         // edge id
  const int f   = gid & 63;                         // feature id
  const int src = edge_index[e];
  const int dst = edge_index[E_REAL + e];
  const int g   = src >> 7;                         // src / N
  const int flat = (g << 14) + ((src - (g << 7)) << 7) + (dst - (g << 7));
  atomicAdd(out_val + (size_t)flat * OUT_DIM + f, edge_attr[gid]);
}

// ---------------------------------------------------------------------------
extern "C" void kernel_launch(void* const* d_in, const int* in_sizes, int n_in,
                              void* d_out, int out_size, void* d_ws, size_t ws_size,
                              hipStream_t stream) {
  // inputs (setup_inputs order): batch, edge_index, edge_attr, gm_index, gm_val, W
  const int*   edge_index = (const int*)d_in[1];
  const float* edge_attr  = (const float*)d_in[2];
  const float* gm_val     = (const float*)d_in[4];
  const float* W          = (const float*)d_in[5];

  // outputs concatenated flat: out_idx [2*TOTAL] then out_val [TOTAL*OUT]
  float* out_idx = (float*)d_out;
  float* out_val = (float*)d_out + 2 * (size_t)TOTAL;

  gm_idx_kernel<<<TOTAL / 256, 256, 0, stream>>>(out_idx);
  gm_gemm_kernel<<<TOTAL / (16 * 8), 256, 0, stream>>>(gm_val, W, out_val);
  gm_scatter_kernel<<<(E_REAL * OUT_DIM) / 256, 256, 0, stream>>>(
      edge_index, edge_attr, out_val);
}